// GNNModel_7791070675006
// MI455X (gfx1250) — compile-verified
//
#include <hip/hip_runtime.h>
#include <math.h>

// ---------------------------------------------------------------------------
// 2-layer GraphConv (DGL norm='both') for gfx1250 / MI455X.
//   layer: h = (x * outdeg^-1/2) @ W ; agg[dst] += h[src] ; y = agg*indeg^-1/2 + b
// Dense transforms run on V_WMMA_F32_16X16X4_F32 (exact f32 MMA, wave32).
// Whole working set (~40 MB) is L2-resident (192 MB); scatter-add uses
// global_atomic_add_f32 which resolves at L2.
// ---------------------------------------------------------------------------

#define NN 50000      // nodes
#define NE 800000     // edges
#define FI 64         // in feats
#define FH 64         // hidden
#define FO 32         // classes

typedef __attribute__((ext_vector_type(2))) float v2f;
typedef __attribute__((ext_vector_type(8))) float v8f;

// ---- workspace layout (floats) --------------------------------------------
#define NPAD 50048                            // NN rounded up to 64
#define OFF_ONORM 0                           // [NPAD] out-deg -> out-norm
#define OFF_INORM (NPAD)                      // [NPAD] in-deg  -> in-norm
#define OFF_H1    (2*NPAD)                    // [NN*64] layer-1 transformed feats
#define OFF_AGG1  (OFF_H1 + NN*FH)            // [NN*64] layer-1 aggregate
#define OFF_H2    (OFF_AGG1 + NN*FH)          // [NN*32] layer-2 transformed feats
#define OFF_AGG2  (OFF_H2 + NN*FO)            // [NN*32] layer-2 aggregate
#define WS_FLOATS (OFF_AGG2 + NN*FO)          // 9,700,096 floats (~38.8 MB)

// ---------------------------------------------------------------------------
__global__ void k_zero(float4* p, int n4) {
    int i = blockIdx.x * blockDim.x + threadIdx.x;
    if (i < n4) p[i] = make_float4(0.f, 0.f, 0.f, 0.f);
}

// one thread per edge: accumulate degrees
__global__ void k_degrees(const int* __restrict__ src, const int* __restrict__ dst,
                          float* __restrict__ outdeg, float* __restrict__ indeg, int ne) {
    int e = blockIdx.x * blockDim.x + threadIdx.x;
    if (e < ne) {
        atomicAdd(&outdeg[src[e]], 1.0f);
        atomicAdd(&indeg[dst[e]], 1.0f);
    }
}

// in-place: deg -> clip(deg,1)^-1/2  (covers both norm arrays contiguously)
__global__ void k_norms(float* __restrict__ d, int n) {
    int i = blockIdx.x * blockDim.x + threadIdx.x;
    if (i < n) d[i] = rsqrtf(fmaxf(d[i], 1.0f));
}

// ---------------------------------------------------------------------------
// WMMA GEMM:  Out[N x NCOLS] = preop(A[N x 64]) @ W[64 x NCOLS]
//   MODE 0: preop(row r) = A[r,:] * rowsc[r]                       (layer 1)
//   MODE 1: preop(row r) = relu(A[r,:] * rowsc[r] + bias[:])       (layer 2,
//           fuses layer-1 epilogue into layer-2 A-staging)
// Block covers 32 rows; waves = 2 row-tiles x (NCOLS/16) col-tiles.
// ---------------------------------------------------------------------------
template <int NCOLS, int MODE>
__global__ __launch_bounds__(64 * (NCOLS / 16))
void k_gemm_wmma(const float* __restrict__ A, const float* __restrict__ W,
                 const float* __restrict__ rowsc, const float* __restrict__ bias,
                 float* __restrict__ Out, int nRows) {
    constexpr int NT  = NCOLS / 16;   // col tiles
    constexpr int NTH = 64 * NT;      // threads per block
    __shared__ float lA[32][68];            // pad: stride 68 % 64 banks = 4 -> conflict-free column reads
    __shared__ float lB[64][NCOLS + 4];

    const int tid = threadIdx.x;
    const int rowbase = blockIdx.x * 32;

    // ---- stage W (64 x NCOLS), coalesced float4 ----
    {
        constexpr int TOT4 = 64 * NCOLS / 4;
        constexpr int PER  = TOT4 / NTH;
#pragma unroll
        for (int i = 0; i < PER; ++i) {
            int idx = tid + i * NTH;
            int r = idx / (NCOLS / 4), c4 = idx % (NCOLS / 4);
            float4 w4 = ((const float4*)(W + r * NCOLS))[c4];
            float* p = &lB[r][c4 * 4];
            p[0] = w4.x; p[1] = w4.y; p[2] = w4.z; p[3] = w4.w;
        }
    }
    // ---- stage A rows (32 x 64) with fused per-row pre-op ----
    {
        constexpr int TOT4 = 32 * 64 / 4;               // 512 float4
        constexpr int PER  = (TOT4 + NTH - 1) / NTH;
#pragma unroll
        for (int i = 0; i < PER; ++i) {
            int idx = tid + i * NTH;
            if (idx < TOT4) {
                int r = idx / 16, c4 = idx % 16;
                int grow = rowbase + r;
                float4 v = make_float4(0.f, 0.f, 0.f, 0.f);
                if (grow < nRows) {
                    v = ((const float4*)(A + (size_t)grow * 64))[c4];
                    float s = rowsc[grow];
                    if (MODE == 0) {
                        v.x *= s; v.y *= s; v.z *= s; v.w *= s;
                    } else {
                        v.x = fmaxf(fmaf(v.x, s, bias[c4 * 4 + 0]), 0.f);
                        v.y = fmaxf(fmaf(v.y, s, bias[c4 * 4 + 1]), 0.f);
                        v.z = fmaxf(fmaf(v.z, s, bias[c4 * 4 + 2]), 0.f);
                        v.w = fmaxf(fmaf(v.w, s, bias[c4 * 4 + 3]), 0.f);
                    }
                }
                float* p = &lA[r][c4 * 4];
                p[0] = v.x; p[1] = v.y; p[2] = v.z; p[3] = v.w;
            }
        }
    }
    __syncthreads();

    // ---- per-wave 16x16 tile, K-loop of 16 steps of K=4 ----
    const int wid  = tid >> 5;
    const int lane = tid & 31;
    const int rt   = wid / NT;         // row tile 0..1
    const int ct   = wid % NT;         // col tile
    const int m    = lane & 15;
    const int kh   = (lane < 16) ? 0 : 2;   // ISA 7.12.2: A 16x4 / B 4x16 half-wave K split
    const int arow = rt * 16 + m;
    const int bcol = ct * 16 + m;

    v8f acc = {};
#pragma unroll
    for (int k4 = 0; k4 < 16; ++k4) {
        const int kb = k4 * 4 + kh;
        v2f a, b;
        a.x = lA[arow][kb];
        a.y = lA[arow][kb + 1];
        b.x = lB[kb][bcol];
        b.y = lB[kb + 1][bcol];
        // (neg_a, A, neg_b, B, c_mod, C, reuse_a, reuse_b)
        acc = __builtin_amdgcn_wmma_f32_16x16x4_f32(false, a, false, b,
                                                    (short)0, acc, false, false);
    }

    // ---- store C tile (VGPR r: lanes0-15 -> M=r, lanes16-31 -> M=r+8) ----
    const int rext = (lane < 16) ? 0 : 8;
#pragma unroll
    for (int r = 0; r < 8; ++r) {
        int grow = rowbase + rt * 16 + r + rext;
        if (grow < nRows) Out[(size_t)grow * NCOLS + ct * 16 + m] = acc[r];
    }
}

// ---------------------------------------------------------------------------
// Edge scatter: one wave per edge. 64-feat: lane reads float2 (coalesced
// 256 B gather of h[src]) and does 2 f32 atomics into agg[dst].
// ---------------------------------------------------------------------------
__global__ void k_scatter64(const float* __restrict__ h, const int* __restrict__ src,
                            const int* __restrict__ dst, float* __restrict__ agg, int ne) {
    int e = (blockIdx.x * blockDim.x + threadIdx.x) >> 5;
    int lane = threadIdx.x & 31;
    if (e >= ne) return;
    int s = src[e], d = dst[e];
    float2 v = ((const float2*)(h + (size_t)s * 64))[lane];
    atomicAdd(&agg[(size_t)d * 64 + 2 * lane + 0], v.x);
    atomicAdd(&agg[(size_t)d * 64 + 2 * lane + 1], v.y);
}

// 32-feat: one wave per edge, one float per lane
__global__ void k_scatter32(const float* __restrict__ h, const int* __restrict__ src,
                            const int* __restrict__ dst, float* __restrict__ agg, int ne) {
    int e = (blockIdx.x * blockDim.x + threadIdx.x) >> 5;
    int lane = threadIdx.x & 31;
    if (e >= ne) return;
    int s = src[e], d = dst[e];
    float v = h[(size_t)s * 32 + lane];
    atomicAdd(&agg[(size_t)d * 32 + lane], v);
}

// out = agg2 * in_norm[row] + b2[col]
__global__ void k_finalize(const float* __restrict__ agg2, const float* __restrict__ innorm,
                           const float* __restrict__ b2, float* __restrict__ out, int n) {
    int i = blockIdx.x * blockDim.x + threadIdx.x;
    if (i < n) {
        int row = i >> 5, c = i & 31;
        out[i] = fmaf(agg2[i], innorm[row], b2[c]);
    }
}

// ---------------------------------------------------------------------------
extern "C" void kernel_launch(void* const* d_in, const int* in_sizes, int n_in,
                              void* d_out, int out_size, void* d_ws, size_t ws_size,
                              hipStream_t stream) {
    const float* features = (const float*)d_in[0];
    const float* W1       = (const float*)d_in[1];
    const float* b1       = (const float*)d_in[2];
    const float* W2       = (const float*)d_in[3];
    const float* b2       = (const float*)d_in[4];
    const int*   src      = (const int*)d_in[5];
    const int*   dst      = (const int*)d_in[6];
    float*       out      = (float*)d_out;

    float* ws     = (float*)d_ws;
    float* onorm  = ws + OFF_ONORM;
    float* inorm  = ws + OFF_INORM;
    float* h1     = ws + OFF_H1;
    float* agg1   = ws + OFF_AGG1;
    float* h2     = ws + OFF_H2;
    float* agg2   = ws + OFF_AGG2;

    const int nNodes = in_sizes[0] / FI;   // 50000
    const int nEdges = in_sizes[5];        // 800000

    // 0) zero scratch (norm accumulators + aggregates; h1/h2 fully overwritten)
    {
        int n4 = WS_FLOATS / 4;
        k_zero<<<(n4 + 255) / 256, 256, 0, stream>>>((float4*)d_ws, n4);
    }
    // 1) degrees (shared by both layers) -> 2) norms
    k_degrees<<<(nEdges + 255) / 256, 256, 0, stream>>>(src, dst, onorm, inorm, nEdges);
    k_norms<<<(2 * NPAD + 255) / 256, 256, 0, stream>>>(onorm, 2 * NPAD);

    const int gemmBlocks = (nNodes + 31) / 32;

    // 3) h1 = (feat * onorm) @ W1          [WMMA f32 16x16x4]
    k_gemm_wmma<FH, 0><<<gemmBlocks, 256, 0, stream>>>(features, W1, onorm, nullptr, h1, nNodes);
    // 4) agg1[dst] += h1[src]
    k_scatter64<<<(nEdges * 32 + 255) / 256, 256, 0, stream>>>(h1, src, dst, agg1, nEdges);
    // 5) h2 = relu(agg1 * inorm + b1) @ W2  (layer-1 epilogue fused into A-staging)
    k_gemm_wmma<FO, 1><<<gemmBlocks, 128, 0, stream>>>(agg1, W2, inorm, b1, h2, nNodes);
    // 6) agg2[dst] += h2[src]
    k_scatter32<<<(nEdges * 32 + 255) / 256, 256, 0, stream>>>(h2, src, dst, agg2, nEdges);
    // 7) out = agg2 * inorm + b2
    k_finalize<<<(nNodes * FO + 255) / 256, 256, 0, stream>>>(agg2, inorm, b2, out, nNodes * FO);
}